// CustomAttention_26431228739592
// MI455X (gfx1250) — compile-verified
//
#include <hip/hip_runtime.h>

// CustomAttention on gfx1250 (MI455X):
//   Phase 1: Q = X*Wq^T + bq ; K = X*Wk^T + bk   via V_WMMA_F32_16X16X4_F32
//   Phase 2: per-edge gather + dot(q[src], k[dst]) / sqrt(128), wave32 reduce
//
// FP32 WMMA is the right precision: workload is memory/gather bound
// (~135 MB HBM traffic, 6.6 GFLOP), and Q+K (51.2 MB) are L2-resident (192 MB).

typedef __attribute__((ext_vector_type(2))) float v2f;
typedef __attribute__((ext_vector_type(8))) float v8f;

#define IN_DIM 256
#define W_DIM 128
#define LDS_STRIDE 260   // 256 + 4 pad -> bank = (4*m + c) % 64, conflict-free

__global__ __launch_bounds__(256) void qk_gemm_wmma(
    const float* __restrict__ X,
    const float* __restrict__ Wq, const float* __restrict__ bq,
    const float* __restrict__ Wk, const float* __restrict__ bk,
    float* __restrict__ Q, float* __restrict__ K)
{
    __shared__ float As[16 * LDS_STRIDE];

    const int tid = threadIdx.x;
    const long r0 = (long)blockIdx.x * 16;   // 16-row block of X

    // Stage 16x256 fp32 A-panel into LDS; 256 threads x 4 float4 each,
    // fully coalesced on global, 16B-aligned on LDS (stride 1040B % 16 == 0).
    #pragma unroll
    for (int j = 0; j < 4; ++j) {
        int flat = (j * 256 + tid) * 4;     // float index 0..4095
        int row  = flat >> 8;               // /256
        int col  = flat & 255;
        float4 v = *(const float4*)(X + (r0 + row) * IN_DIM + col);
        *(float4*)(&As[row * LDS_STRIDE + col]) = v;
    }
    __syncthreads();

    const int wave = tid >> 5;          // 8 waves, one 16-col tile each
    const int lane = tid & 31;
    const int half = lane >> 4;         // 0: lanes 0-15, 1: lanes 16-31
    const int mn   = lane & 15;         // M for A-frag, N for B/C/D frags
    const int col0 = wave * 16;

    v8f accq = {};
    v8f acck = {};

    const float* arow  = &As[mn * LDS_STRIDE];
    const float* wqrow = Wq + (long)(col0 + mn) * IN_DIM;
    const float* wkrow = Wk + (long)(col0 + mn) * IN_DIM;

    // K loop: 64 steps of 4; A frag shared between Q and K accumulation.
    #pragma unroll 4
    for (int k0 = 0; k0 < IN_DIM; k0 += 4) {
        const int kk = k0 + 2 * half;   // even -> 8B-aligned v2f loads
        v2f a  = *(const v2f*)(arow  + kk);   // A 16x4: lanes0-15 K0..1, lanes16-31 K2..3
        v2f bqf = *(const v2f*)(wqrow + kk);  // B 4x16: N across lanes, same K split
        v2f bkf = *(const v2f*)(wkrow + kk);
        accq = __builtin_amdgcn_wmma_f32_16x16x4_f32(
                   false, a, false, bqf, (short)0, accq, false, false);
        acck = __builtin_amdgcn_wmma_f32_16x16x4_f32(
                   false, a, false, bkf, (short)0, acck, false, false);
    }

    // Bias depends only on output column N = col0 + mn (same for all 8 regs).
    const float biasq = bq[col0 + mn];
    const float biask = bk[col0 + mn];

    // C/D layout: VGPR i -> row (i + 8*half), col = col0 + mn
    #pragma unroll
    for (int i = 0; i < 8; ++i) {
        const long row = r0 + i + 8 * half;
        Q[row * W_DIM + col0 + mn] = accq[i] + biasq;
        K[row * W_DIM + col0 + mn] = acck[i] + biask;
    }
}

__global__ __launch_bounds__(256) void edge_scores(
    const float* __restrict__ Q, const float* __restrict__ K,
    const long long* __restrict__ ei, float* __restrict__ out, long long E)
{
    const long long e = ((long long)blockIdx.x * blockDim.x + threadIdx.x) >> 5;
    const int lane = threadIdx.x & 31;
    if (e >= E) return;                      // wave-uniform branch

    const long long s = ei[e];               // src node
    const long long d = ei[E + e];           // dst node

    // 32 lanes x float4 = 128 floats: one coalesced 512B row read each.
    const float4 qv = ((const float4*)(Q + s * W_DIM))[lane];
    const float4 kv = ((const float4*)(K + d * W_DIM))[lane];
    float p = qv.x * kv.x + qv.y * kv.y + qv.z * kv.z + qv.w * kv.w;

    // wave32 butterfly reduction
    #pragma unroll
    for (int off = 16; off > 0; off >>= 1)
        p += __shfl_xor(p, off, 32);

    if (lane == 0)
        out[e] = p * 0.08838834764831845f;   // 1/sqrt(128)
}

extern "C" void kernel_launch(void* const* d_in, const int* in_sizes, int n_in,
                              void* d_out, int out_size, void* d_ws, size_t ws_size,
                              hipStream_t stream) {
    const float*     X  = (const float*)d_in[0];       // [N, 256]
    const float*     Wq = (const float*)d_in[1];       // [128, 256]
    const float*     bq = (const float*)d_in[2];       // [128]
    const float*     Wk = (const float*)d_in[3];       // [128, 256]
    const float*     bk = (const float*)d_in[4];       // [128]
    const long long* ei = (const long long*)d_in[5];   // [2, E] int64

    const int       N = in_sizes[0] / IN_DIM;          // 50000 (multiple of 16)
    const long long E = in_sizes[5] / 2;               // 1600000

    float* Q = (float*)d_ws;                           // N*128 fp32
    float* K = Q + (size_t)N * W_DIM;                  // N*128 fp32

    qk_gemm_wmma<<<N / 16, 256, 0, stream>>>(X, Wq, bq, Wk, bk, Q, K);

    const int eblocks = (int)((E * 32 + 255) / 256);   // one wave per edge
    edge_scores<<<eblocks, 256, 0, stream>>>(Q, K, ei, (float*)d_out, E);
}